// VariationalCircuit_52578989638123
// MI455X (gfx1250) — compile-verified
//
#include <hip/hip_runtime.h>

// ---------------------------------------------------------------------------
// 23-qubit variational circuit simulator for gfx1250 (MI455X).
//  - Per layer: 23 fused 1q unitaries (RZ*RY*RX), then CNOT ladder.
//  - CNOT ladder == prefix-XOR index permutation, folded into writeback.
//  - 4-qubit groups -> 16x16 complex unitaries applied with
//    V_WMMA_F32_16X16X4_F32 (full fp32 matrix pipeline).
//  - Tile fill uses GLOBAL_LOAD_ASYNC_TO_LDS_B128 (ASYNCcnt) -- CDNA5 async
//    copy path, no VGPR staging.
//  - 2-bit XOR swizzle on LDS DWORD index kills bank conflicts for the p=0
//    group while keeping 16B chunks contiguous (B128-safe).
// ---------------------------------------------------------------------------

#define DIMQ (1u << 23)
constexpr int NQ = 23;
constexpr int TILE_BITS = 14;
constexpr int TILE = 1 << TILE_BITS;       // 16384 amps, 128 KB LDS (re+im)
constexpr int THREADS = 256;               // 8 waves of 32
constexpr int NWAVES = THREADS / 32;
constexpr int NBLK = (int)(DIMQ >> TILE_BITS);  // 512 workgroups per sweep

typedef float v2f __attribute__((ext_vector_type(2)));
typedef float v8f __attribute__((ext_vector_type(8)));

__device__ __forceinline__ float2 cmulf(float2 a, float2 b) {
  return make_float2(a.x * b.x - a.y * b.y, a.x * b.y + a.y * b.x);
}

// y_k = x_0 ^ x_1 ^ ... ^ x_k  (the full CNOT ladder as an index map)
__device__ __forceinline__ unsigned prefix_xor23(unsigned x) {
  unsigned y = x;
  y ^= y << 1; y ^= y << 2; y ^= y << 4; y ^= y << 8; y ^= y << 16;
  return y & (DIMQ - 1u);
}

// Insert group index g (4 bits) at local bit position p of "other" index n.
__device__ __forceinline__ int posOf(int g, int n, int p) {
  int low = n & ((1 << p) - 1);
  return ((n >> p) << (p + 4)) | (g << p) | low;
}

// Bank-conflict swizzle on LDS DWORD index: XOR bits [3:2] with bits [7:6].
// Bijective, keeps 16B (4-DWORD) chunks contiguous, and makes the stride-16
// access pattern of the p=0 group hit 16 distinct banks.
__device__ __forceinline__ unsigned swz(unsigned dw) {
  return dw ^ (((dw >> 6) & 3u) << 2);
}

// ---------------------------------------------------------------------------
// Build the 14 fused 16x16 complex group unitaries from params.
//   gi 0..3  (sweep A, qubits 0..13):  {0,1,2,3} {4,5,6,7} {8,9,10,11} {I,I,12,13}
//   gi 4..6  (sweep B, qubits 14..22): {14,15,16,17} {18,19,20,21} {I,I,I,22}
// ---------------------------------------------------------------------------
__global__ __launch_bounds__(THREADS) void qsim_build_mats(
    const float* __restrict__ params, float2* __restrict__ mats) {
  __shared__ float2 U[2][NQ][2][2];
  const int tid = threadIdx.x;
  if (tid < 2 * NQ) {
    const int layer = tid / NQ, q = tid % NQ;
    const float* p = params + (layer * NQ + q) * 3;
    float cx = __cosf(0.5f * p[0]), sx = __sinf(0.5f * p[0]);
    float cy = __cosf(0.5f * p[1]), sy = __sinf(0.5f * p[1]);
    float cz = __cosf(0.5f * p[2]), sz = __sinf(0.5f * p[2]);
    // Uy*Ux:
    float2 a00 = make_float2( cy * cx,  sy * sx);
    float2 a01 = make_float2(-sy * cx, -cy * sx);
    float2 a10 = make_float2( sy * cx, -cy * sx);
    float2 a11 = make_float2( cy * cx, -sy * sx);
    // Uz = diag(e^{-i tz/2}, e^{+i tz/2})
    float2 e0 = make_float2(cz, -sz), e1 = make_float2(cz, sz);
    U[layer][q][0][0] = cmulf(e0, a00);
    U[layer][q][0][1] = cmulf(e0, a01);
    U[layer][q][1][0] = cmulf(e1, a10);
    U[layer][q][1][1] = cmulf(e1, a11);
  }
  __syncthreads();
  const int r = tid >> 4, c = tid & 15;
  const int table[7][4] = {{0, 1, 2, 3},     {4, 5, 6, 7},
                           {8, 9, 10, 11},   {-1, -1, 12, 13},
                           {14, 15, 16, 17}, {18, 19, 20, 21},
                           {-1, -1, -1, 22}};
  for (int m = 0; m < 14; ++m) {
    const int layer = m / 7, gi = m % 7;
    float2 prod = make_float2(1.f, 0.f);
    bool zero = false;
    #pragma unroll
    for (int k = 0; k < 4; ++k) {
      const int q = table[gi][k];
      const int rb = (r >> k) & 1, cb = (c >> k) & 1;
      if (q < 0) {
        if (rb != cb) zero = true;
      } else {
        prod = cmulf(prod, U[layer][q][rb][cb]);
      }
    }
    mats[m * 256 + r * 16 + c] = zero ? make_float2(0.f, 0.f) : prod;
  }
}

// ---------------------------------------------------------------------------
// One sweep: async-load a 2^14 tile into LDS, apply 16x16 complex unitaries
// on 4-bit groups via V_WMMA_F32_16X16X4_F32, write back.
//  sweepB==0: tile = contiguous (local bits = qubits 0..13), groups p=0,4,8,10
//  sweepB==1: tile local bits = qubits {0..4} U {14..22}, groups p=5,9,10,
//             writeback goes through the prefix-XOR (CNOT-ladder) permutation.
// ---------------------------------------------------------------------------
__global__ __launch_bounds__(THREADS) void qsim_sweep(
    const float* __restrict__ srcRe, const float* __restrict__ srcIm,
    float* __restrict__ dstRe, float* __restrict__ dstIm,
    const float2* __restrict__ mats, const int sweepB) {
  __shared__ float sRe[TILE];
  __shared__ float sIm[TILE];
  const int tid = threadIdx.x;
  const unsigned blk = blockIdx.x;

  // ---- async tile fill: GLOBAL_LOAD_ASYNC_TO_LDS_B128, 16B per lane ------
  // Sweep A: 512B contiguous global run per wave-instruction; sweep B: 128B runs.
  {
    const unsigned ldsRe0 = (unsigned)(size_t)(&sRe[0]);
    const unsigned ldsIm0 = (unsigned)(size_t)(&sIm[0]);
    #pragma unroll 1
    for (int k = 0; k < TILE / 4 / THREADS; ++k) {          // 16 uniform iters
      const unsigned j = (unsigned)(tid + k * THREADS);     // float4 chunk id
      const unsigned x4 = sweepB
          ? (((j & 7u) << 2) | (blk << 5) | ((j >> 3) << 14))
          : (blk * (unsigned)TILE + 4u * j);
      const unsigned ldsOff = swz(4u * j) * 4u;             // chunk-safe swizzle
      const unsigned long long gre = (unsigned long long)(size_t)(srcRe + x4);
      const unsigned long long gim = (unsigned long long)(size_t)(srcIm + x4);
      asm volatile("global_load_async_to_lds_b128 %0, %1, off"
                   :: "v"(ldsRe0 + ldsOff), "v"(gre) : "memory");
      asm volatile("global_load_async_to_lds_b128 %0, %1, off"
                   :: "v"(ldsIm0 + ldsOff), "v"(gim) : "memory");
    }
  }
  asm volatile("s_wait_asynccnt 0" ::: "memory");

  const int ngroups = sweepB ? 3 : 4;
  const int gpArr[4] = {sweepB ? 5 : 0, sweepB ? 9 : 4, sweepB ? 10 : 8, 10};
  const int lane = tid & 31;
  const int wave = tid >> 5;
  const int half = (lane >> 4) & 1;   // 0: lanes 0-15, 1: lanes 16-31
  const int nlo = lane & 15;

  for (int grp = 0; grp < ngroups; ++grp) {
    __syncthreads();
    const int p = gpArr[grp];
    const float2* __restrict__ M = mats + grp * 256;

    // A operands: unitary columns, ISA A-layout (K = 4c + {0,1} / {2,3})
    v2f ar[4], ai[4], nai[4];
    #pragma unroll
    for (int c4 = 0; c4 < 4; ++c4) {
      const int col = 4 * c4 + 2 * half;
      const float2 m0 = M[nlo * 16 + col];
      const float2 m1 = M[nlo * 16 + col + 1];
      ar[c4]  = v2f{ m0.x,  m1.x};
      ai[c4]  = v2f{ m0.y,  m1.y};
      nai[c4] = v2f{-m0.y, -m1.y};
    }

    // 1024 "other" columns per group, 16 per WMMA tile, 64 tiles / 8 waves.
    // Uniform trip count (8): no exec-masked loop; EXEC stays all-ones for WMMA.
    #pragma unroll 1
    for (int it = 0; it < (TILE >> 8) / NWAVES; ++it) {
      const int t = wave + it * NWAVES;
      const int n = t * 16 + nlo;
      v2f br[4], bi[4];
      #pragma unroll
      for (int c4 = 0; c4 < 4; ++c4) {
        const int k0 = 4 * c4 + 2 * half;
        const unsigned px = swz((unsigned)posOf(k0, n, p));
        const unsigned py = swz((unsigned)posOf(k0 + 1, n, p));
        br[c4] = v2f{sRe[px], sRe[py]};
        bi[c4] = v2f{sIm[px], sIm[py]};
      }
      v8f accRe = {0, 0, 0, 0, 0, 0, 0, 0};
      v8f accIm = {0, 0, 0, 0, 0, 0, 0, 0};
      // D_re = Ur*Sr - Ui*Si ; D_im = Ur*Si + Ui*Sr  (16 f32 WMMAs)
      #pragma unroll
      for (int c4 = 0; c4 < 4; ++c4)
        accRe = __builtin_amdgcn_wmma_f32_16x16x4_f32(
            false, ar[c4], false, br[c4], (short)0, accRe, false, false);
      #pragma unroll
      for (int c4 = 0; c4 < 4; ++c4)
        accRe = __builtin_amdgcn_wmma_f32_16x16x4_f32(
            false, nai[c4], false, bi[c4], (short)0, accRe, false, false);
      #pragma unroll
      for (int c4 = 0; c4 < 4; ++c4)
        accIm = __builtin_amdgcn_wmma_f32_16x16x4_f32(
            false, ar[c4], false, bi[c4], (short)0, accIm, false, false);
      #pragma unroll
      for (int c4 = 0; c4 < 4; ++c4)
        accIm = __builtin_amdgcn_wmma_f32_16x16x4_f32(
            false, ai[c4], false, br[c4], (short)0, accIm, false, false);
      // D layout: VGPR r -> M = r (+8 for upper lane half)
      #pragma unroll
      for (int rr = 0; rr < 8; ++rr) {
        const unsigned pm = swz((unsigned)posOf(rr + 8 * half, n, p));
        sRe[pm] = accRe[rr];
        sIm[pm] = accIm[rr];
      }
    }
  }
  __syncthreads();

  // ---- writeback; sweep B applies the whole-layer CNOT-ladder permutation
  #pragma unroll 1
  for (int k = 0; k < TILE / THREADS; ++k) {                // 64 uniform iters
    const int i = tid + k * THREADS;
    const unsigned g = sweepB
        ? ((unsigned)(i & 31) | (blk << 5) | ((unsigned)(i >> 5) << 14))
        : (blk * (unsigned)TILE + (unsigned)i);
    const unsigned y = sweepB ? prefix_xor23(g) : g;
    const unsigned si = swz((unsigned)i);
    dstRe[y] = sRe[si];
    dstIm[y] = sIm[si];
  }
}

// ---------------------------------------------------------------------------
extern "C" void kernel_launch(void* const* d_in, const int* in_sizes, int n_in,
                              void* d_out, int out_size, void* d_ws,
                              size_t ws_size, hipStream_t stream) {
  const float* stateRe = (const float*)d_in[0];            // (2, DIM) f32
  const float* stateIm = stateRe + DIMQ;
  const float* params  = (const float*)d_in[1];            // 138 f32
  float* outRe = (float*)d_out;
  float* outIm = outRe + DIMQ;
  float* wsRe = (float*)d_ws;                              // 64 MB ping-pong
  float* wsIm = wsRe + DIMQ;
  float2* mats = (float2*)(wsRe + 2 * (size_t)DIMQ);       // 14 * 2 KB

  qsim_build_mats<<<1, THREADS, 0, stream>>>(params, mats);

  // layer 0
  qsim_sweep<<<NBLK, THREADS, 0, stream>>>(stateRe, stateIm, outRe, outIm,
                                           mats + 0 * 256, 0);
  qsim_sweep<<<NBLK, THREADS, 0, stream>>>(outRe, outIm, wsRe, wsIm,
                                           mats + 4 * 256, 1);
  // layer 1
  qsim_sweep<<<NBLK, THREADS, 0, stream>>>(wsRe, wsIm, wsRe, wsIm,
                                           mats + 7 * 256, 0);
  qsim_sweep<<<NBLK, THREADS, 0, stream>>>(wsRe, wsIm, outRe, outIm,
                                           mats + 11 * 256, 1);
}